// HeteroLinkGCN_69990787056126
// MI455X (gfx1250) — compile-verified
//
#include <hip/hip_runtime.h>

typedef __attribute__((ext_vector_type(2))) float v2f;
typedef __attribute__((ext_vector_type(8))) float v8f;

#define TPB 256

#if defined(__has_builtin)
#if __has_builtin(__builtin_amdgcn_sched_group_barrier)
// per k-step: 2 global A prefetches, NDS ds_load_b64 B fragments, NW WMMAs
#define SCHED_KSTEP(NDS, NW)                              \
  __builtin_amdgcn_sched_group_barrier(0x020, 2, 0);      \
  __builtin_amdgcn_sched_group_barrier(0x100, (NDS), 0);  \
  __builtin_amdgcn_sched_group_barrier(0x008, (NW), 0);
#endif
#endif
#ifndef SCHED_KSTEP
#define SCHED_KSTEP(NDS, NW)
#endif

__device__ __forceinline__ v8f wmma4(v2f a, v2f b, v8f c) {
  return __builtin_amdgcn_wmma_f32_16x16x4_f32(false, a, false, b, (short)0, c,
                                               false, false);
}

__global__ void zero_f32(float* __restrict__ p, int n) {
  int i = blockIdx.x * blockDim.x + threadIdx.x;
  if (i < n) p[i] = 0.0f;
}

__global__ void degree_count(const int* __restrict__ dst, float* __restrict__ deg, int E) {
  int e = blockIdx.x * blockDim.x + threadIdx.x;
  if (e < E) atomicAdd(&deg[dst[e]], 1.0f);
}

// x[M,256] @ {W0,W1,W2}[256,64] -> {out0,out1,out2}; bias applies to out0 only.
// MIX: out0 = 0.5*(acc + bias + x[:, :64])  (word-node epilogue, NMAT==1)
// NT = 16-col tiles per wave (4: full width; 2: wave pairs split N so the
// 3-matrix variant stays at 96 accumulator VGPRs -> no VGPR-MSB mode).
// LDS W image is pair-interleaved + bank-rotated:
//   (k,n) -> (k>>1)*128 + ((2n + ((k>>1)&1)*32) & 127) + (k&1)
// so a B fragment is one aligned ds_load_b64 AND lanes 0-15 / 16-31 (which
// read adjacent K-pair blocks, since (p&1)==laneHi when k%4==0) land on
// disjoint bank halves -> conflict-free. sched_group_barrier forces all
// NMAT*NT B loads to issue before the WMMA burst (staggered dscnt waits).
template <int NMAT, int NT, int MIX>
__global__ __launch_bounds__(TPB)
__attribute__((amdgpu_waves_per_eu(1, 4)))
void gemm_wmma(
    const float* __restrict__ x,
    const float* __restrict__ W0, const float* __restrict__ W1,
    const float* __restrict__ W2, const float* __restrict__ bias,
    float* __restrict__ out0, float* __restrict__ out1, float* __restrict__ out2,
    int M)
{
  extern __shared__ float wLds[];
  const int tid = threadIdx.x;

  #pragma unroll
  for (int m = 0; m < NMAT; ++m) {
    const float* W = (m == 0) ? W0 : ((m == 1) ? W1 : W2);
    for (int rem = tid; rem < 16384; rem += TPB) {
      int k = rem >> 6, n = rem & 63;
      int p = k >> 1;
      int pos = (((n << 1) + ((p & 1) << 5)) & 127) + (k & 1);
      wLds[(m << 14) + (p << 7) + pos] = W[rem];
    }
  }
  __syncthreads();

  const int wave   = tid >> 5;
  const int lane   = tid & 31;
  const int laneLo = lane & 15;
  const int laneHi = lane >> 4;
  const int koff   = laneHi << 1;            // lanes 0-15: K k..k+1; 16-31: k+2..k+3
  const int rowWave = (NT == 4) ? wave : (wave >> 1);
  const int ntBase  = (NT == 4) ? 0 : ((wave & 1) * 2);          // col-tile base
  const int rowWavesPerBlk = (NT == 4) ? 8 : 4;
  const int tileRow = (blockIdx.x * rowWavesPerBlk + rowWave) << 5; // 32 rows/wave
  if (tileRow >= M) return;                  // wave-uniform: EXEC all-ones for WMMA

  int r1 = tileRow + 16 + laneLo;
  if (r1 >= M) r1 = M - 1;                   // clamp tail loads; stores are guarded
  const float* xr0 = x + (size_t)(tileRow + laneLo) * 256;
  const float* xr1 = x + (size_t)r1 * 256;

  // per-lane invariant LDS read offsets (rotation folded in: (p&1)==laneHi)
  int posA[NT];
  #pragma unroll
  for (int nt = 0; nt < NT; ++nt)
    posA[nt] = (((ntBase + nt + laneHi) & 3) << 5) + (laneLo << 1);
  const int laneBlk = laneHi << 7;           // odd K-pair block for lanes 16-31

  v8f acc[NMAT][2][NT] = {};

  v2f a0 = *(const v2f*)&xr0[koff];
  v2f a1 = *(const v2f*)&xr1[koff];
  for (int k = 0; k < 252; k += 4) {
    v2f na0 = *(const v2f*)&xr0[k + 4 + koff];   // prefetch next k-step
    v2f na1 = *(const v2f*)&xr1[k + 4 + koff];
    const int kb = ((k >> 1) << 7) + laneBlk;
    v2f b[NMAT][NT];
    #pragma unroll
    for (int m = 0; m < NMAT; ++m)
      #pragma unroll
      for (int nt = 0; nt < NT; ++nt)
        b[m][nt] = *(const v2f*)&wLds[(m << 14) + kb + posA[nt]];
    #pragma unroll
    for (int m = 0; m < NMAT; ++m)
      #pragma unroll
      for (int nt = 0; nt < NT; ++nt) {
        acc[m][0][nt] = wmma4(a0, b[m][nt], acc[m][0][nt]);
        acc[m][1][nt] = wmma4(a1, b[m][nt], acc[m][1][nt]);
      }
    a0 = na0; a1 = na1;
    SCHED_KSTEP(NMAT * NT, 2 * NMAT * NT)
  }
  {
    const int kb = ((252 >> 1) << 7) + laneBlk;  // final k-step
    v2f b[NMAT][NT];
    #pragma unroll
    for (int m = 0; m < NMAT; ++m)
      #pragma unroll
      for (int nt = 0; nt < NT; ++nt)
        b[m][nt] = *(const v2f*)&wLds[(m << 14) + kb + posA[nt]];
    #pragma unroll
    for (int m = 0; m < NMAT; ++m)
      #pragma unroll
      for (int nt = 0; nt < NT; ++nt) {
        acc[m][0][nt] = wmma4(a0, b[m][nt], acc[m][0][nt]);
        acc[m][1][nt] = wmma4(a1, b[m][nt], acc[m][1][nt]);
      }
  }

  // C/D layout: VGPR r -> (row = r + 8*laneHi, col = laneLo) per 16x16 tile
  #pragma unroll
  for (int m = 0; m < NMAT; ++m) {
    float* out = (m == 0) ? out0 : ((m == 1) ? out1 : out2);
    #pragma unroll
    for (int t = 0; t < 2; ++t) {
      #pragma unroll
      for (int nt = 0; nt < NT; ++nt) {
        #pragma unroll
        for (int r = 0; r < 8; ++r) {
          int orow = tileRow + t * 16 + r + (laneHi << 3);
          if (orow < M) {
            int ocol = (ntBase + nt) * 16 + laneLo;
            float v = acc[m][t][nt][r];
            if (m == 0) {
              if (bias) v += bias[ocol];
              if (MIX)  v = 0.5f * (v + x[(size_t)orow * 256 + ocol]);
            }
            out[(size_t)orow * 64 + ocol] = v;
          }
        }
      }
    }
  }
}

// 16 threads per edge, float4 per thread: out[dst] += (coef*w/deg[dst]) * H[src]
__global__ void scatter_edges(const int* __restrict__ src, const int* __restrict__ dst,
                              const float* __restrict__ w, const float* __restrict__ H,
                              const float* __restrict__ deg, float* __restrict__ out,
                              int E, float coef)
{
  long long t = (long long)blockIdx.x * blockDim.x + threadIdx.x;
  int e = (int)(t >> 4);
  if (e >= E) return;
  int c = ((int)t & 15) << 2;
  int s = src[e], d = dst[e];
  float scale = coef * w[e] / deg[d];       // deg >= 1 for any dst with an edge
  const float4 h = *(const float4*)&H[(size_t)s * 64 + c];
  float* o = out + (size_t)d * 64 + c;
  atomicAdd(o + 0, scale * h.x);
  atomicAdd(o + 1, scale * h.y);
  atomicAdd(o + 2, scale * h.z);
  atomicAdd(o + 3, scale * h.w);
}

extern "C" void kernel_launch(void* const* d_in, const int* in_sizes, int n_in,
                              void* d_out, int out_size, void* d_ws, size_t ws_size,
                              hipStream_t stream)
{
  const float* x_ac       = (const float*)d_in[0];
  const float* x_w        = (const float*)d_in[1];
  const int*   src_tic    = (const int*)d_in[2];
  const int*   dst_tic    = (const int*)d_in[3];
  const float* w_tic      = (const float*)d_in[4];
  const int*   src_rel    = (const int*)d_in[5];
  const int*   dst_rel    = (const int*)d_in[6];
  const float* w_rel      = (const float*)d_in[7];
  const float* W_self_tic = (const float*)d_in[8];
  const float* W_neigh_tic= (const float*)d_in[9];
  const float* b_tic      = (const float*)d_in[10];
  const float* W_self_rel = (const float*)d_in[11];
  const float* W_neigh_rel= (const float*)d_in[12];
  const float* b_rel      = (const float*)d_in[13];

  const int NA    = in_sizes[0] / 256;
  const int NW    = in_sizes[1] / 256;
  const int E_TIC = in_sizes[2];
  const int E_REL = in_sizes[5];

  float* out_ac = (float*)d_out;
  float* out_w  = (float*)d_out + (size_t)NA * 64;

  auto cdiv = [](long long a, long long b) { return (int)((a + b - 1) / b); };
  const int tilesA = cdiv(NA, 32);
  const int tilesW = cdiv(NW, 32);

  const bool fused3 = ws_size >= ((size_t)NA * 64 * 2 + NA) * sizeof(float);

  if (fused3) {
    // one x_ac sweep: self -> out_ac, H1 = x_ac@W_neigh_tic, H2 = x_ac@W_neigh_rel
    float* H1  = (float*)d_ws;
    float* H2  = H1 + (size_t)NA * 64;
    float* deg = H2 + (size_t)NA * 64;

    zero_f32<<<cdiv(NA, TPB), TPB, 0, stream>>>(deg, NA);
    degree_count<<<cdiv(E_TIC, TPB), TPB, 0, stream>>>(dst_tic, deg, E_TIC);
    gemm_wmma<3, 2, 0><<<cdiv(tilesA, 4), TPB, 3 * 16384 * sizeof(float), stream>>>(
        x_ac, W_self_tic, W_neigh_tic, W_neigh_rel, b_tic, out_ac, H1, H2, NA);
    scatter_edges<<<cdiv((long long)E_TIC * 16, TPB), TPB, 0, stream>>>(
        src_tic, dst_tic, w_tic, H1, deg, out_ac, E_TIC, 1.0f);

    zero_f32<<<cdiv(NW, TPB), TPB, 0, stream>>>(deg, NW);
    degree_count<<<cdiv(E_REL, TPB), TPB, 0, stream>>>(dst_rel, deg, E_REL);
    gemm_wmma<1, 4, 1><<<cdiv(tilesW, 8), TPB, 16384 * sizeof(float), stream>>>(
        x_w, W_self_rel, nullptr, nullptr, b_rel, out_w, nullptr, nullptr, NW);
    scatter_edges<<<cdiv((long long)E_REL * 16, TPB), TPB, 0, stream>>>(
        src_rel, dst_rel, w_rel, H2, deg, out_w, E_REL, 0.5f);
  } else {
    // low-workspace path: single H buffer reused across relations
    float* H   = (float*)d_ws;
    float* deg = H + (size_t)NA * 64;

    zero_f32<<<cdiv(NA, TPB), TPB, 0, stream>>>(deg, NA);
    degree_count<<<cdiv(E_TIC, TPB), TPB, 0, stream>>>(dst_tic, deg, E_TIC);
    gemm_wmma<2, 4, 0><<<cdiv(tilesA, 8), TPB, 2 * 16384 * sizeof(float), stream>>>(
        x_ac, W_self_tic, W_neigh_tic, nullptr, b_tic, out_ac, H, nullptr, NA);
    scatter_edges<<<cdiv((long long)E_TIC * 16, TPB), TPB, 0, stream>>>(
        src_tic, dst_tic, w_tic, H, deg, out_ac, E_TIC, 1.0f);

    zero_f32<<<cdiv(NW, TPB), TPB, 0, stream>>>(deg, NW);
    degree_count<<<cdiv(E_REL, TPB), TPB, 0, stream>>>(dst_rel, deg, E_REL);
    gemm_wmma<1, 4, 0><<<cdiv(tilesA, 8), TPB, 16384 * sizeof(float), stream>>>(
        x_ac, W_neigh_rel, nullptr, nullptr, nullptr, H, nullptr, nullptr, NA);
    gemm_wmma<1, 4, 1><<<cdiv(tilesW, 8), TPB, 16384 * sizeof(float), stream>>>(
        x_w, W_self_rel, nullptr, nullptr, b_rel, out_w, nullptr, nullptr, NW);
    scatter_edges<<<cdiv((long long)E_REL * 16, TPB), TPB, 0, stream>>>(
        src_rel, dst_rel, w_rel, H, deg, out_w, E_REL, 0.5f);
  }
}